// InfoNCE_62904091018137
// MI455X (gfx1250) — compile-verified
//
#include <hip/hip_runtime.h>
#include <math.h>

// CDNA5 fp32 WMMA operand types: A/B of 16x16x4 f32 = 2 VGPRs, C/D = 8 VGPRs.
typedef __attribute__((ext_vector_type(2))) float v2f;
typedef __attribute__((ext_vector_type(8))) float v8f;

#define B_SZ 512
#define D_SZ 174
#define D_PAD 176   // K padded to multiple of 4 for 16x16x4 WMMA

// ---------------------------------------------------------------------------
// Kernel 1: per-row prep. mask = (true>=0), tn = normalize(true*mask),
// p = pred (zero-padded), p2 = pred^2. One block per row.
// ---------------------------------------------------------------------------
__global__ __launch_bounds__(256) void prep_kernel(
    const float* __restrict__ t, const float* __restrict__ pr,
    float* __restrict__ tn, float* __restrict__ mk,
    float* __restrict__ p, float* __restrict__ p2)
{
    __shared__ float red[256];
    const int r = blockIdx.x;   // row
    const int d = threadIdx.x;  // element

    float tv = 0.0f, m = 0.0f, pv = 0.0f;
    if (d < D_SZ) {
        tv = t[r * D_SZ + d];
        pv = pr[r * D_SZ + d];
        m  = (tv >= 0.0f) ? 1.0f : 0.0f;
    }
    const float tm = tv * m;

    red[d] = tm * tm;
    __syncthreads();
    for (int s = 128; s > 0; s >>= 1) {
        if (d < s) red[d] += red[d + s];
        __syncthreads();
    }
    const float inv = 1.0f / fmaxf(sqrtf(red[0]), 1e-12f);

    if (d < D_PAD) {
        tn[r * D_PAD + d] = tm * inv;
        mk[r * D_PAD + d] = m;
        p [r * D_PAD + d] = pv;
        p2[r * D_PAD + d] = pv * pv;
    }
}

// ---------------------------------------------------------------------------
// Kernel 2: dual fp32 WMMA GEMM (num = P @ Tn^T, den = P^2 @ M^T) fused with
// spectral-angle transform.
//
// Block = 8 waves, all sharing the same output row-tile ti; wave w computes
// column tile tj = (blockIdx&3)*8 + w. The A operands (pred / pred^2 rows of
// tile ti, 2 x 16 x 176 floats = 22528 B) are staged once per block into LDS
// via the CDNA5 async global->LDS path (GLOBAL_LOAD_ASYNC_TO_LDS_B128,
// ASYNCcnt), giving 8x reuse; B operands are per-wave global b64 loads.
//
// A layout (16x4 f32): lane<16 -> row=lane, K=k0/k0+1 ; lane>=16 -> K=k0+2/+3.
// B layout (4x16 f32): mirrored, N = lane&15.
// D layout: VGPR v -> row = v + 8*(lane>=16), col = lane&15.
// ---------------------------------------------------------------------------
__global__ __launch_bounds__(256) void gemm_sa_kernel(
    const float* __restrict__ p,  const float* __restrict__ p2,
    const float* __restrict__ tn, const float* __restrict__ mk,
    float* __restrict__ S)
{
    // [0 .. 16*D_PAD) = pred tile, [16*D_PAD .. 2*16*D_PAD) = pred^2 tile
    __shared__ __align__(16) float smem[2 * 16 * D_PAD];   // 22528 B

    const int lane = threadIdx.x & 31;
    const int wave = threadIdx.x >> 5;
    const int ti = blockIdx.x >> 2;                 // shared row tile
    const int tj = ((blockIdx.x & 3) << 3) + wave;  // per-wave col tile

    // ---- async stage of A tiles (pred, pred^2) into LDS --------------------
    {
        const char* gsrc_p  = (const char*)(p  + (size_t)ti * 16 * D_PAD);
        const char* gsrc_p2 = (const char*)(p2 + (size_t)ti * 16 * D_PAD);
        // low 32 bits of a generic pointer into LDS == wave-relative LDS addr
        const unsigned lds_p  = (unsigned)(uintptr_t)(&smem[0]);
        const unsigned lds_p2 = (unsigned)(uintptr_t)(&smem[16 * D_PAD]);
        // 11264 B per matrix = 22 chunks of 512 B (32 lanes x 16 B)
        for (int c = wave; c < 22; c += 8) {
            const unsigned off = (unsigned)c * 512u + (unsigned)lane * 16u;
            unsigned long long ga  = (unsigned long long)(uintptr_t)(gsrc_p  + off);
            unsigned long long ga2 = (unsigned long long)(uintptr_t)(gsrc_p2 + off);
            unsigned la  = lds_p  + off;
            unsigned la2 = lds_p2 + off;
            asm volatile("global_load_async_to_lds_b128 %0, %1, off"
                         :: "v"(la),  "v"(ga)  : "memory");
            asm volatile("global_load_async_to_lds_b128 %0, %1, off"
                         :: "v"(la2), "v"(ga2) : "memory");
        }
        asm volatile("s_wait_asynccnt 0x0" ::: "memory");
        __syncthreads();
    }

    const int mrow = lane & 15;
    const int koff = (lane >> 4) << 1;       // 0 for lanes 0-15, 2 for 16-31

    const float* sA  = smem + mrow * D_PAD + koff;               // LDS
    const float* sA2 = smem + 16 * D_PAD + mrow * D_PAD + koff;  // LDS
    const float* pB  = tn + (size_t)(tj * 16 + mrow) * D_PAD + koff;
    const float* pB2 = mk + (size_t)(tj * 16 + mrow) * D_PAD + koff;

    v8f cn = {};  // numerator accumulator
    v8f cd = {};  // denominator accumulator

#pragma unroll 4
    for (int k0 = 0; k0 < D_PAD; k0 += 4) {
        v2f a  = *(const v2f*)(sA  + k0);     // ds_load_b64
        v2f b  = *(const v2f*)(pB  + k0);     // global_load_b64
        v2f a2 = *(const v2f*)(sA2 + k0);
        v2f b2 = *(const v2f*)(pB2 + k0);
        cn = __builtin_amdgcn_wmma_f32_16x16x4_f32(
            false, a,  false, b,  (short)0, cn, false, false);
        cd = __builtin_amdgcn_wmma_f32_16x16x4_f32(
            false, a2, false, b2, (short)0, cd, false, false);
    }

    const int col   = tj * 16 + (lane & 15);
    const int rbase = ti * 16 + ((lane >> 4) << 3);
#pragma unroll
    for (int v = 0; v < 8; ++v) {
        const int row = rbase + v;
        float num = cn[v];
        float den = cd[v];
        float dn  = sqrtf(fmaxf(den, 0.0f));
        float re  = num / fmaxf(dn, 1e-12f);
        re = fminf(1.0f, fmaxf(-1.0f, re));
        float sa = 1.0f - (2.0f / 3.14159265358979323846f) * acosf(re);
        S[(size_t)row * B_SZ + col] = sa * 10.0f;  // / TEMP (=0.1)
    }
}

// ---------------------------------------------------------------------------
// Kernel 3: per-row logsumexp; rowloss[i] = S[i][i] - lse(S[i][:]).
// One block (256 threads) per row; each thread covers 2 of 512 columns.
// ---------------------------------------------------------------------------
__global__ __launch_bounds__(256) void row_lse_kernel(
    const float* __restrict__ S, float* __restrict__ rowloss)
{
    __shared__ float red[256];
    const int i = blockIdx.x;
    const int t = threadIdx.x;

    const float a = S[(size_t)i * B_SZ + t];
    const float b = S[(size_t)i * B_SZ + 256 + t];

    red[t] = fmaxf(a, b);
    __syncthreads();
    for (int s = 128; s > 0; s >>= 1) {
        if (t < s) red[t] = fmaxf(red[t], red[t + s]);
        __syncthreads();
    }
    const float m = red[0];
    __syncthreads();

    red[t] = expf(a - m) + expf(b - m);
    __syncthreads();
    for (int s = 128; s > 0; s >>= 1) {
        if (t < s) red[t] += red[t + s];
        __syncthreads();
    }
    if (t == 0) {
        const float lse = m + logf(red[0]);
        rowloss[i] = S[(size_t)i * B_SZ + i] - lse;
    }
}

// ---------------------------------------------------------------------------
// Kernel 4: loss = -mean(rowloss). Single block; fully overwrites d_out.
// ---------------------------------------------------------------------------
__global__ __launch_bounds__(256) void final_kernel(
    const float* __restrict__ rowloss, float* __restrict__ out)
{
    __shared__ float red[256];
    const int t = threadIdx.x;
    red[t] = rowloss[t] + rowloss[t + 256];
    __syncthreads();
    for (int s = 128; s > 0; s >>= 1) {
        if (t < s) red[t] += red[t + s];
        __syncthreads();
    }
    if (t == 0) out[0] = -red[0] / (float)B_SZ;
}

extern "C" void kernel_launch(void* const* d_in, const int* in_sizes, int n_in,
                              void* d_out, int out_size, void* d_ws, size_t ws_size,
                              hipStream_t stream)
{
    (void)in_sizes; (void)n_in; (void)out_size; (void)ws_size;

    const float* t  = (const float*)d_in[0];  // true  [512,174]
    const float* pr = (const float*)d_in[1];  // pred  [512,174]

    // Workspace layout (floats): all offsets 16B-aligned (row stride 176*4=704B).
    float* ws = (float*)d_ws;
    float* tn = ws;                          // 512*176  normalized masked true
    float* mk = tn + B_SZ * D_PAD;           // 512*176  mask
    float* p  = mk + B_SZ * D_PAD;           // 512*176  pred (padded)
    float* p2 = p  + B_SZ * D_PAD;           // 512*176  pred^2
    float* S  = p2 + B_SZ * D_PAD;           // 512*512  nce scores
    float* rl = S  + B_SZ * B_SZ;            // 512      per-row losses

    prep_kernel   <<<B_SZ, 256, 0, stream>>>(t, pr, tn, mk, p, p2);
    gemm_sa_kernel<<<128,  256, 0, stream>>>(p, p2, tn, mk, S);
    row_lse_kernel<<<B_SZ, 256, 0, stream>>>(S, rl);
    final_kernel  <<<1,    256, 0, stream>>>(rl, (float*)d_out);
}